// MultiHeadAttention_16767552324052
// MI455X (gfx1250) — compile-verified
//
#include <hip/hip_runtime.h>
#include <hip/hip_bf16.h>
#include <math.h>

// Problem constants (from reference)
#define L 32768
#define D 512
#define H 4
#define HD 128

typedef float v2f __attribute__((ext_vector_type(2)));
typedef float v8f __attribute__((ext_vector_type(8)));

// ---------------------------------------------------------------------------
// Kernel 1: q = Wq x0 + bq ; w[h][d] = scale * sum_j q[h*128+j] * Wk[512+h*128+j][d]
// One block of 512 threads. Tiny (2 x 1MB weight reads).
// ---------------------------------------------------------------------------
__global__ void k1_make_w(const float* __restrict__ x,
                          const float* __restrict__ W,   // in_proj_weight [1536,512]
                          const float* __restrict__ b,   // in_proj_bias [1536]
                          float* __restrict__ wvec)      // [4*512]
{
    __shared__ float qs[D];
    const int i = threadIdx.x;            // 0..511
    // q[i] = dot(W row i, x[0]) + b[i]
    const float* wr = W + (size_t)i * D;
    float acc = b[i];
    for (int d = 0; d < D; ++d) acc += wr[d] * x[d];
    qs[i] = acc;
    __syncthreads();
    // w[h][i] = scale * sum_j qs[h*128+j] * Wk[h*128+j][i]   (Wk = rows 512..1023)
    const float scale = 0.08838834764831845f;   // 1/sqrt(128)
    for (int h = 0; h < H; ++h) {
        float s = 0.0f;
        const float* col = W + (size_t)(D + h * HD) * D + i;   // column i, coalesced
        for (int j = 0; j < HD; ++j) s += qs[h * HD + j] * col[(size_t)j * D];
        wvec[h * D + i] = s * scale;
    }
}

// ---------------------------------------------------------------------------
// Kernel 2: scores s[l][h] = w_h . x[l]  via V_WMMA_F32_16X16X4_F32.
// A = w (heads 0..3, rows 4..15 zero) from padded LDS; B = 16 rows of x / wave.
// Block 256 (8 waves) -> 128 rows/block, grid 256. Streams x once (64 MB).
// ---------------------------------------------------------------------------
#define WPAD 516   // 512 + 4 floats pad: stride%64 = 4 banks -> conflict-free b64 loads
__global__ void k2_scores(const float* __restrict__ x,
                          const float* __restrict__ wvec,
                          float* __restrict__ s)          // [L][4]
{
    __shared__ __align__(16) float wl[16 * WPAD];
    const int tid = threadIdx.x;
    for (int i = tid; i < 16 * WPAD; i += 256) wl[i] = 0.0f;
    __syncthreads();
    for (int i = tid; i < H * D; i += 256) wl[(i >> 9) * WPAD + (i & 511)] = wvec[i];
    __syncthreads();

    const int lane  = tid & 31;
    const int half2 = (lane >> 4) << 1;     // 0 or 2 : K sub-pair per lane half
    const int n     = lane & 15;            // row-within-tile (N) / head-row (M)
    const int wave  = tid >> 5;
    const int row   = blockIdx.x * 128 + wave * 16 + n;

    const float* xr = x  + (size_t)row * D + half2;
    const float* wr = wl + n * WPAD + half2;

    v8f acc = {0.f, 0.f, 0.f, 0.f, 0.f, 0.f, 0.f, 0.f};
#pragma unroll 8
    for (int k = 0; k < D; k += 4) {
        v2f a = *(const v2f*)(wr + k);      // ds_load_b64  (A: 16x4 of w, heads in M)
        v2f bb = *(const v2f*)(xr + k);     // global_load_b64 (B: 4x16 of x, rows in N)
        acc = __builtin_amdgcn_wmma_f32_16x16x4_f32(
            false, a, false, bb, (short)0, acc, false, false);
    }
    // D tile: VGPR h, lanes 0-15 hold head h for rows n (heads 4..15 are zero)
    if (lane < 16) {
        float4 o;
        o.x = acc[0]; o.y = acc[1]; o.z = acc[2]; o.w = acc[3];
        *(float4*)(s + (size_t)row * 4) = o;
    }
}

// ---------------------------------------------------------------------------
// Kernel 3: per-head softmax stats: stats[h] = max, stats[4+h] = 1/sum(exp).
// grid 4 (one block per head), block 256. 512 KB of s, negligible.
// ---------------------------------------------------------------------------
__global__ void k3_stats(const float* __restrict__ s, float* __restrict__ stats)
{
    __shared__ float red[256];
    const int h = blockIdx.x, tid = threadIdx.x;
    float m = -3.4e38f;
    for (int l = tid; l < L; l += 256) m = fmaxf(m, s[l * 4 + h]);
    red[tid] = m; __syncthreads();
    for (int off = 128; off > 0; off >>= 1) {
        if (tid < off) red[tid] = fmaxf(red[tid], red[tid + off]);
        __syncthreads();
    }
    m = red[0]; __syncthreads();
    float sum = 0.0f;
    for (int l = tid; l < L; l += 256) sum += expf(s[l * 4 + h] - m);
    red[tid] = sum; __syncthreads();
    for (int off = 128; off > 0; off >>= 1) {
        if (tid < off) red[tid] += red[tid + off];
        __syncthreads();
    }
    if (tid == 0) { stats[h] = m; stats[4 + h] = 1.0f / red[0]; }
}

// ---------------------------------------------------------------------------
// Kernel 4: per-block partial y: ypart[blk][h][c] = sum over 128 rows of
// attn[h,l] * x[l][c]. Streams x a second time (L2-resident: 64MB < 192MB L2).
// Deterministic: no atomics; grid 256, block 256 (each thread owns 2 columns).
// ---------------------------------------------------------------------------
__global__ void k4_weighted(const float* __restrict__ x,
                            const float* __restrict__ s,
                            const float* __restrict__ stats,
                            float* __restrict__ ypart)     // [256][4*512]
{
    __shared__ __align__(16) float p[128 * 4];
    __shared__ float ms[8];
    const int tid = threadIdx.x;
    if (tid < 8) ms[tid] = stats[tid];
    __syncthreads();
    const int rowbase = blockIdx.x * 128;
    for (int i = tid; i < 128 * 4; i += 256) {
        const int h = i & 3;
        p[i] = expf(s[(size_t)(rowbase + (i >> 2)) * 4 + h] - ms[h]) * ms[4 + h];
    }
    __syncthreads();

    float acc[8] = {0.f, 0.f, 0.f, 0.f, 0.f, 0.f, 0.f, 0.f};
    const int c0 = tid, c1 = tid + 256;
    for (int l = 0; l < 128; ++l) {
        const float* xr = x + (size_t)(rowbase + l) * D;
        const float x0 = xr[c0], x1 = xr[c1];            // coalesced
        const float4 pl = *(const float4*)(p + l * 4);   // LDS broadcast
        acc[0] += pl.x * x0; acc[1] += pl.y * x0; acc[2] += pl.z * x0; acc[3] += pl.w * x0;
        acc[4] += pl.x * x1; acc[5] += pl.y * x1; acc[6] += pl.z * x1; acc[7] += pl.w * x1;
    }
    float* yp = ypart + (size_t)blockIdx.x * (H * D);
    for (int h = 0; h < H; ++h) {
        yp[h * D + c0] = acc[h];
        yp[h * D + c1] = acc[4 + h];
    }
}

// ---------------------------------------------------------------------------
// Kernel 5: deterministic reduce of 256 partials -> y[4*512]. grid 8, block 256.
// ---------------------------------------------------------------------------
__global__ void k5_reduce(const float* __restrict__ ypart, float* __restrict__ y)
{
    const int g = blockIdx.x * 256 + threadIdx.x;   // 0..2047
    float sum = 0.0f;
    for (int b = 0; b < 256; ++b) sum += ypart[(size_t)b * (H * D) + g];
    y[g] = sum;
}

// ---------------------------------------------------------------------------
// Kernel 6: o[i] = Wv row (1024+i) . y[head(i)] + bv[1024+i];
//           out[i] = Wout row i . o + bout[i].  One block, 512 threads.
// ---------------------------------------------------------------------------
__global__ void k6_out(const float* __restrict__ W,    // in_proj_weight
                       const float* __restrict__ b,    // in_proj_bias
                       const float* __restrict__ Wo,   // out_proj_weight
                       const float* __restrict__ bo,   // out_proj_bias
                       const float* __restrict__ y,
                       float* __restrict__ out)
{
    __shared__ float o[D];
    const int i = threadIdx.x;         // 0..511
    const int h = i >> 7;
    const float* wr = W + (size_t)(2 * D + i) * D;
    const float* yh = y + h * D;
    float acc = b[2 * D + i];
    for (int d = 0; d < D; ++d) acc += wr[d] * yh[d];
    o[i] = acc;
    __syncthreads();
    const float* wor = Wo + (size_t)i * D;
    float acc2 = bo[i];
    for (int e = 0; e < D; ++e) acc2 += wor[e] * o[e];
    out[i] = acc2;
}

// ---------------------------------------------------------------------------
// Launch. Workspace layout (floats):
//   s:      [0, 131072)         scores [L][4]
//   w:      [131072, 133120)    head vectors [4][512]
//   stats:  [133120, 133128)    max[4], invsum[4]
//   y:      [133128, 135176)    reduced y [4][512]
//   ypart:  [135176, 659464)    per-block partials [256][2048]
// total ~2.64 MB.
// ---------------------------------------------------------------------------
extern "C" void kernel_launch(void* const* d_in, const int* in_sizes, int n_in,
                              void* d_out, int out_size, void* d_ws, size_t ws_size,
                              hipStream_t stream)
{
    const float* x   = (const float*)d_in[0];
    const float* W   = (const float*)d_in[1];
    const float* b   = (const float*)d_in[2];
    const float* Wo  = (const float*)d_in[3];
    const float* bo  = (const float*)d_in[4];
    float* out = (float*)d_out;

    float* ws    = (float*)d_ws;
    float* s     = ws;
    float* wvec  = ws + 131072;
    float* stats = ws + 133120;
    float* y     = ws + 133128;
    float* ypart = ws + 135176;

    hipLaunchKernelGGL(k1_make_w,  dim3(1),   dim3(512), 0, stream, x, W, b, wvec);
    hipLaunchKernelGGL(k2_scores,  dim3(256), dim3(256), 0, stream, x, wvec, s);
    hipLaunchKernelGGL(k3_stats,   dim3(4),   dim3(256), 0, stream, s, stats);
    hipLaunchKernelGGL(k4_weighted,dim3(256), dim3(256), 0, stream, x, s, stats, ypart);
    hipLaunchKernelGGL(k5_reduce,  dim3(8),   dim3(256), 0, stream, ypart, y);
    hipLaunchKernelGGL(k6_out,     dim3(1),   dim3(512), 0, stream, W, b, Wo, bo, y, out);
}